// MPC_86715389706590
// MI455X (gfx1250) — compile-verified
//
#include <hip/hip_runtime.h>

// ---------------------------------------------------------------------------
// Batched LQR (T=64, B=512, N=32, M=16, D=48) for gfx1250 (MI455X).
// One workgroup (4 wave32) per batch element; Riccati recursion in LDS;
// dense products on V_WMMA_F32_16X16X4_F32; C/F streamed with
// GLOBAL_LOAD_ASYNC_TO_LDS_B128 (ASYNCcnt), K/k spilled with
// GLOBAL_STORE_ASYNC_FROM_LDS_B128.
// ---------------------------------------------------------------------------

#define T_ 64
#define B_ 512
#define N_ 32
#define M_ 16
#define D_ 48   // N + M

typedef float v2f __attribute__((ext_vector_type(2)));
typedef float v8f __attribute__((ext_vector_type(8)));
typedef int   v4i __attribute__((vector_size(16)));   // matches builtin pointee

// ---- async global<->LDS path (gfx1250), with synchronous fallback ---------
#if defined(__has_builtin)
#if __has_builtin(__builtin_amdgcn_global_load_async_to_lds_b128)
#define ASYNC_LD 1
#endif
#if __has_builtin(__builtin_amdgcn_global_store_async_from_lds_b128)
#define ASYNC_ST 1
#endif
#endif
#ifndef ASYNC_LD
#define ASYNC_LD 0
#endif
#ifndef ASYNC_ST
#define ASYNC_ST 0
#endif

static __device__ __forceinline__ void g2l_b128(const float* g, float* l) {
#if ASYNC_LD
  __builtin_amdgcn_global_load_async_to_lds_b128(
      (__attribute__((address_space(1))) v4i*)g,
      (__attribute__((address_space(3))) v4i*)l, 0, 0);
#else
  *(float4*)l = *(const float4*)g;
#endif
}

static __device__ __forceinline__ void l2g_b128(float* g, const float* l) {
#if ASYNC_ST
  __builtin_amdgcn_global_store_async_from_lds_b128(
      (__attribute__((address_space(1))) v4i*)g,
      (__attribute__((address_space(3))) v4i*)l, 0, 0);
#else
  *(float4*)g = *(const float4*)l;
#endif
}

static __device__ __forceinline__ void wait_async() {
#if ASYNC_LD || ASYNC_ST
#if defined(__has_builtin) && __has_builtin(__builtin_amdgcn_s_wait_asynccnt)
  __builtin_amdgcn_s_wait_asynccnt(0);
#else
  asm volatile("s_wait_asynccnt 0" ::: "memory");
#endif
#endif
}

// ---- WMMA 16x16 f32 tile helpers (layouts per CDNA5 ISA 7.12.2) -----------
static __device__ __forceinline__ v8f wmma_acc_load(const float* S, int ld, int r0, int c0) {
  const int lane = threadIdx.x & 31, half = lane >> 4, n = lane & 15;
  v8f c;
#pragma unroll
  for (int r = 0; r < 8; ++r) c[r] = S[(r0 + r + 8 * half) * ld + c0 + n];
  return c;
}

static __device__ __forceinline__ void wmma_store(float* S, int ld, int r0, int c0, v8f d) {
  const int lane = threadIdx.x & 31, half = lane >> 4, n = lane & 15;
#pragma unroll
  for (int r = 0; r < 8; ++r) S[(r0 + r + 8 * half) * ld + c0 + n] = d[r];
}

// acc += A(16xK) * B(Kx16); A read (optionally transposed) from LDS row-major.
template <bool TRA>
static __device__ __forceinline__ v8f wmma_mac(const float* A, int lda, int ar,
                                               const float* B, int ldb, int bc,
                                               int K, v8f acc) {
  const int lane = threadIdx.x & 31, half = lane >> 4, l = lane & 15;
#pragma unroll 4
  for (int k0 = 0; k0 < K; k0 += 4) {
    const int ka = k0 + 2 * half;     // A vgpr0: K=0/2, vgpr1: K=1/3 per half
    v2f a, b;
    if (TRA) {                        // logical A = storage^T
      a.x = A[(ka + 0) * lda + ar + l];
      a.y = A[(ka + 1) * lda + ar + l];
    } else {
      a.x = A[(ar + l) * lda + ka + 0];
      a.y = A[(ar + l) * lda + ka + 1];
    }
    b.x = B[(ka + 0) * ldb + bc + l];
    b.y = B[(ka + 1) * ldb + bc + l];
    acc = __builtin_amdgcn_wmma_f32_16x16x4_f32(false, a, false, b,
                                                (short)0, acc, false, false);
  }
  return acc;
}

__global__ __launch_bounds__(128) void
MPC_86715389706590_kernel(const float* __restrict__ x_init,
                          const float* __restrict__ C,
                          const float* __restrict__ c,
                          const float* __restrict__ F,
                          const float* __restrict__ f,
                          float* __restrict__ out,
                          float* __restrict__ ws) {
  const int b    = blockIdx.x;          // batch problem
  const int tid  = threadIdx.x;
  const int wave = tid >> 5;
  const int lane = tid & 31;

  __shared__ __align__(16) float sF[N_ * D_];     // F_t        (32 x 48)
  __shared__ __align__(16) float sQ[D_ * D_];     // C_t -> Q   (48 x 48)
  __shared__ __align__(16) float sV[N_ * N_];     // V          (32 x 32)
  __shared__ __align__(16) float sFtV[D_ * N_];   // F^T V      (48 x 32)
  __shared__ __align__(16) float sK[M_ * N_];     // K          (16 x 32)
  __shared__ __align__(16) float sW[M_ * N_];     // Q_uu K (+Q_ux) (16x32)
  __shared__ float sAug[M_ * 32];                 // [Q_uu | I] Gauss-Jordan
  __shared__ float sNI[M_ * M_];                  // -Q_uu^{-1}
  __shared__ float svv[N_];                       // v
  __shared__ float sq[D_];                        // q
  __shared__ float scv[D_], sfv[N_];              // c_t, f_t
  __shared__ __align__(16) float skv[M_], st2[M_]; // k, q_u + Q_uu k
  __shared__ float sx[D_];                        // tau = [x; u]

  // V0 = 0, v0 = 0
  for (int i = tid; i < N_ * N_; i += 128) sV[i] = 0.f;
  if (tid < N_) svv[tid] = 0.f;

  // ------------------------- backward Riccati pass -------------------------
  for (int t = T_ - 1; t >= 0; --t) {
    const size_t tb = (size_t)t * B_ + b;
    const float* Cp = C + tb * (D_ * D_);
    const float* Fp = F + tb * (N_ * D_);
    // stage C_t -> sQ, F_t -> sF directly into LDS (ASYNCcnt path)
    for (int i = tid; i < (D_ * D_) / 4; i += 128) g2l_b128(Cp + 4 * i, sQ + 4 * i);
    for (int i = tid; i < (N_ * D_) / 4; i += 128) g2l_b128(Fp + 4 * i, sF + 4 * i);
    if (tid < D_) scv[tid] = c[tb * D_ + tid];
    if (tid < N_) sfv[tid] = f[tb * N_ + tid];
    // Prefetch next step's streams (global_prefetch_b8).
    if (t > 0) {
      const char* pc = (const char*)(Cp - B_ * (D_ * D_));
      const char* pf = (const char*)(Fp - B_ * (N_ * D_));
      if (tid < 72) __builtin_prefetch(pc + tid * 128, 0, 1);
      if (tid < 48) __builtin_prefetch(pf + tid * 128, 0, 1);
    }
    wait_async();
    __syncthreads();

    // FtV = F^T V : 48x32 = (3x2) tiles, K = 32
    for (int tt = wave; tt < 6; tt += 4) {
      const int tr = (tt >> 1) * 16, tc = (tt & 1) * 16;
      v8f acc = {};
      acc = wmma_mac<true>(sF, D_, tr, sV, N_, tc, N_, acc);
      wmma_store(sFtV, N_, tr, tc, acc);
    }
    __syncthreads();

    // Q = C + FtV * F : 48x48 = (3x3) tiles, K = 32 (accumulate onto C in sQ)
    for (int tt = wave; tt < 9; tt += 4) {
      const int tr = (tt / 3) * 16, tc = (tt % 3) * 16;
      v8f acc = wmma_acc_load(sQ, D_, tr, tc);
      acc = wmma_mac<false>(sFtV, N_, tr, sF, D_, tc, N_, acc);
      wmma_store(sQ, D_, tr, tc, acc);
    }
    __syncthreads();

    // waves 2..3: q = c + FtV f + F^T v   |   wave 0: Gauss-Jordan on Q_uu
    if (wave >= 2) {
      const int i = tid - 64;
      if (i < D_) {
        float a = scv[i];
        for (int j = 0; j < N_; ++j) a += sFtV[i * N_ + j] * sfv[j];
        for (int j = 0; j < N_; ++j) a += sF[j * D_ + i] * svv[j];
        sq[i] = a;
      }
    }
    if (wave == 0) {
      // lane owns one column of the augmented 16x32 system [Q_uu | I]
      for (int i = 0; i < M_; ++i)
        sAug[i * 32 + lane] = (lane < 16) ? sQ[(N_ + i) * D_ + (N_ + lane)]
                                          : ((lane - 16 == i) ? 1.f : 0.f);
      for (int p = 0; p < M_; ++p) {
        const float inv = 1.0f / sAug[p * 32 + p];
        sAug[p * 32 + lane] *= inv;
        for (int r = 0; r < M_; ++r)
          if (r != p) {
            const float fm = sAug[r * 32 + p];
            sAug[r * 32 + lane] -= fm * sAug[p * 32 + lane];
          }
      }
      if (lane < 16)
        for (int i = 0; i < M_; ++i) sNI[i * M_ + lane] = -sAug[i * 32 + 16 + lane];
    }
    __syncthreads();

    // K = (-Q_uu^{-1}) Q_ux : 16x32, waves 0..1   |   wave 2: k vector
    if (wave < 2) {
      const int tc = wave * 16;
      v8f acc = {};
      acc = wmma_mac<false>(sNI, M_, 0, sQ + N_ * D_, D_, tc, M_, acc);
      wmma_store(sK, N_, 0, tc, acc);
    }
    if (wave == 2 && lane < M_) {
      float a = 0.f;
      for (int j = 0; j < M_; ++j) a += sNI[lane * M_ + j] * sq[N_ + j];
      skv[lane] = a;
    }
    __syncthreads();

    // waves 2..3: W = Q_uu K   |   tid<16: st2 = q_u + Q_uu k
    if (wave >= 2) {
      const int tc = (wave - 2) * 16;
      v8f acc = {};
      acc = wmma_mac<false>(sQ + N_ * D_ + N_, D_, 0, sK, N_, tc, M_, acc);
      wmma_store(sW, N_, 0, tc, acc);
    }
    if (tid < M_) {
      float a = sq[N_ + tid];
      for (int j = 0; j < M_; ++j) a += sQ[(N_ + tid) * D_ + (N_ + j)] * skv[j];
      st2[tid] = a;
    }
    __syncthreads();

    // W += Q_ux (so V' = Q_xx + Q_xu K + K^T (Q_ux + Q_uu K))
    for (int i = tid; i < M_ * N_; i += 128)
      sW[i] += sQ[(N_ + (i >> 5)) * D_ + (i & 31)];
    __syncthreads();

    // V' : 32x32 = 4 tiles, one per wave
    {
      const int tr = (wave >> 1) * 16, tc = (wave & 1) * 16;
      v8f acc = wmma_acc_load(sQ, D_, tr, tc);                     // Q_xx
      acc = wmma_mac<false>(sQ + N_, D_, tr, sK, N_, tc, M_, acc); // + Q_xu K
      acc = wmma_mac<true>(sK, N_, tr, sW, N_, tc, M_, acc);       // + K^T W
      wmma_store(sV, N_, tr, tc, acc);
    }
    __syncthreads();

    // v' = q_x + Q_xu k + K^T (q_u + Q_uu k) ; spill K,k for forward pass
    if (tid < N_) {
      float a = sq[tid];
      for (int j = 0; j < M_; ++j) a += sQ[tid * D_ + (N_ + j)] * skv[j];
      for (int j = 0; j < M_; ++j) a += sK[j * N_ + tid] * st2[j];
      svv[tid] = a;
    }
    {
      float* wsKk = ws + tb * (M_ * N_ + M_);
      for (int i = tid; i < (M_ * N_) / 4; i += 128)
        l2g_b128(wsKk + 4 * i, sK + 4 * i);
      if (tid == 0) {
        l2g_b128(wsKk + M_ * N_ + 0, skv + 0);
        l2g_b128(wsKk + M_ * N_ + 4, skv + 4);
        l2g_b128(wsKk + M_ * N_ + 8, skv + 8);
        l2g_b128(wsKk + M_ * N_ + 12, skv + 12);
      }
    }
    wait_async();
    __syncthreads();
  }

  // ----------------------------- forward rollout ---------------------------
  if (tid < N_) sx[tid] = x_init[(size_t)b * N_ + tid];
  __syncthreads();

  for (int t = 0; t < T_; ++t) {
    const size_t tb = (size_t)t * B_ + b;
    const float* wsKk = ws + tb * (M_ * N_ + M_);
    for (int i = tid; i < (M_ * N_) / 4; i += 128)
      g2l_b128(wsKk + 4 * i, sK + 4 * i);
    if (tid == 0) {
      g2l_b128(wsKk + M_ * N_ + 0, skv + 0);
      g2l_b128(wsKk + M_ * N_ + 4, skv + 4);
      g2l_b128(wsKk + M_ * N_ + 8, skv + 8);
      g2l_b128(wsKk + M_ * N_ + 12, skv + 12);
    }
    const float* Fp = F + tb * (N_ * D_);
    for (int i = tid; i < (N_ * D_) / 4; i += 128) g2l_b128(Fp + 4 * i, sF + 4 * i);
    if (tid < N_) sfv[tid] = f[tb * N_ + tid];
    wait_async();
    __syncthreads();

    // u = K x + k  -> tau = [x; u]
    if (tid < M_) {
      float a = skv[tid];
      for (int j = 0; j < N_; ++j) a += sK[tid * N_ + j] * sx[j];
      sx[N_ + tid] = a;
    }
    __syncthreads();

    // emit tau, compute x' = F tau + f
    if (tid < D_) out[tb * D_ + tid] = sx[tid];
    float xn = 0.f;
    if (tid < N_) {
      xn = sfv[tid];
      for (int j = 0; j < D_; ++j) xn += sF[tid * D_ + j] * sx[j];
    }
    __syncthreads();
    if (tid < N_) sx[tid] = xn;
    __syncthreads();
  }
}

extern "C" void kernel_launch(void* const* d_in, const int* in_sizes, int n_in,
                              void* d_out, int out_size, void* d_ws, size_t ws_size,
                              hipStream_t stream) {
  (void)in_sizes; (void)n_in; (void)out_size; (void)ws_size;
  const float* x_init = (const float*)d_in[0];
  const float* C      = (const float*)d_in[1];
  const float* c      = (const float*)d_in[2];
  const float* F      = (const float*)d_in[3];
  const float* f      = (const float*)d_in[4];
  float* out = (float*)d_out;
  float* ws  = (float*)d_ws;   // needs T*B*(16*32+16)*4 ≈ 66 MB for K,k spill

  MPC_86715389706590_kernel<<<B_, 128, 0, stream>>>(x_init, C, c, F, f, out, ws);
}